// WeLMMoeSparseMoEBlock_31576599560862
// MI455X (gfx1250) — compile-verified
//
#include <hip/hip_runtime.h>
#include <hip/hip_bf16.h>
#include <math.h>

// ---- problem constants (match reference) ----
#define H_DIM  2048
#define NEXP   8
#define I_MOE_ 1408
#define I_SH_  5632
#define T_TOK  1024   // BS*NT = 2*512

typedef __attribute__((ext_vector_type(2))) float v2f;
typedef __attribute__((ext_vector_type(8))) float v8f;

// D = A(16x4) * B(4x16) + C(16x16), all fp32 (V_WMMA_F32_16X16X4_F32)
static __device__ __forceinline__ v8f wmma4(v2f a, v2f b, v8f c) {
  return __builtin_amdgcn_wmma_f32_16x16x4_f32(false, a, false, b, (short)0, c,
                                               false, false);
}

// -------------------------------------------------------------------------
// Router: logits -> softmax -> top2 -> renorm; sigmoid shared gate;
// builds per-expert (token,slot) pair lists for the sparse expert GEMMs.
// -------------------------------------------------------------------------
__global__ void router_kernel(const float* __restrict__ X,
                              const float* __restrict__ GW,   // [8,H]
                              const float* __restrict__ SGW,  // [1,H]
                              float* __restrict__ topw,       // [T,2]
                              float* __restrict__ sgate,      // [T]
                              int*   __restrict__ counts,     // [8]
                              int*   __restrict__ pairs)      // [8*T]
{
  __shared__ float red[NEXP + 1][256];
  const int t   = blockIdx.x;
  const int tid = threadIdx.x;
  const float* xr = X + (size_t)t * H_DIM;

  float acc[NEXP + 1];
#pragma unroll
  for (int e = 0; e <= NEXP; ++e) acc[e] = 0.f;
  for (int h = tid; h < H_DIM; h += 256) {
    const float xv = xr[h];
#pragma unroll
    for (int e = 0; e < NEXP; ++e) acc[e] += xv * GW[e * H_DIM + h];
    acc[NEXP] += xv * SGW[h];
  }
#pragma unroll
  for (int e = 0; e <= NEXP; ++e) red[e][tid] = acc[e];
  __syncthreads();
  for (int off = 128; off > 0; off >>= 1) {
    if (tid < off) {
#pragma unroll
      for (int e = 0; e <= NEXP; ++e) red[e][tid] += red[e][tid + off];
    }
    __syncthreads();
  }
  if (tid == 0) {
    float p[NEXP];
    float mx = -1e30f;
#pragma unroll
    for (int e = 0; e < NEXP; ++e) mx = fmaxf(mx, red[e][0]);
    float sum = 0.f;
#pragma unroll
    for (int e = 0; e < NEXP; ++e) { p[e] = __expf(red[e][0] - mx); sum += p[e]; }
    const float inv = 1.f / sum;
#pragma unroll
    for (int e = 0; e < NEXP; ++e) p[e] *= inv;
    int i0 = 0;
#pragma unroll
    for (int e = 1; e < NEXP; ++e) if (p[e] > p[i0]) i0 = e;
    int i1 = (i0 == 0) ? 1 : 0;
#pragma unroll
    for (int e = 0; e < NEXP; ++e) if (e != i0 && p[e] > p[i1]) i1 = e;
    const float ws = p[i0] + p[i1];
    topw[t * 2 + 0] = p[i0] / ws;
    topw[t * 2 + 1] = p[i1] / ws;
    sgate[t] = 1.f / (1.f + __expf(-red[NEXP][0]));
    int s0 = atomicAdd(&counts[i0], 1);
    pairs[i0 * T_TOK + s0] = t * 2 + 0;
    int s1 = atomicAdd(&counts[i1], 1);
    pairs[i1 * T_TOK + s1] = t * 2 + 1;
  }
}

// -------------------------------------------------------------------------
// Fused gate_up + SiLU*mul GEMM.  One wave computes a 32(M) x 64(N) tile of
// h = silu(X.Wg^T) * (X.Wu^T).  16 f32 accumulator tiles live in VGPRs; the
// operand registers form a 4-slot, distance-2 software pipeline: loads for
// K-step s+2 are issued while the WMMAs of step s execute, so every WMMA
// burst waits only on the oldest of 2-3 in-flight load groups.
// pairs==nullptr -> dense rows; else gathered pair rows.  K % 16 == 0.
// W layout: [ (e) , 2*Ntot , K ] row-major, g rows then u rows.
// -------------------------------------------------------------------------
__global__ void __launch_bounds__(32, 1)
gateup_silu_wmma(const float* __restrict__ X,
                 const float* __restrict__ W,
                 float* __restrict__ Hout,     // [*, Ntot]
                 const int* __restrict__ pairs,
                 const int* __restrict__ counts,
                 int K, int Ntot)
{
  const int e   = blockIdx.z;
  const int cnt = pairs ? counts[e] : T_TOK;
  const int m_base = blockIdx.y * 32;
  if (m_base >= cnt) return;
  const int* plist = pairs ? (pairs + (size_t)e * T_TOK) : nullptr;
  const float* Wc  = W + (size_t)e * (size_t)(2 * Ntot) * (size_t)K;

  const int lane = threadIdx.x;
  const int m    = lane & 15;
  const int half = lane >> 4;          // 0: K 0,1 / M 0-7 ; 1: K 2,3 / M 8-15
  const int kl   = half << 1;

  int gm0 = m_base + m;      if (gm0 > cnt - 1) gm0 = cnt - 1;
  int gm1 = m_base + 16 + m; if (gm1 > cnt - 1) gm1 = cnt - 1;
  const int ar0 = plist ? (plist[gm0] >> 1) : gm0;
  const int ar1 = plist ? (plist[gm1] >> 1) : gm1;
  const float* ap0 = X + (size_t)ar0 * K + kl;
  const float* ap1 = X + (size_t)ar1 * K + kl;

  const int n0 = blockIdx.x * 64;
  const float* bp[8];
#pragma unroll
  for (int nt = 0; nt < 4; ++nt) {
    const int n = n0 + nt * 16 + m;
    bp[nt]     = Wc + (size_t)n * K + kl;            // gate rows
    bp[nt + 4] = Wc + (size_t)(Ntot + n) * K + kl;   // up rows
  }

  v8f acc[2][8];
  const v8f vzero = {};
#pragma unroll
  for (int mh = 0; mh < 2; ++mh)
#pragma unroll
    for (int j = 0; j < 8; ++j) acc[mh][j] = vzero;

  // 4-slot operand pipeline (distance 2)
  v2f A0[4], A1[4], Bf[4][8];

  auto LD = [&](int k, int s) {
    A0[s] = *(const v2f*)(ap0 + k);
    A1[s] = *(const v2f*)(ap1 + k);
#pragma unroll
    for (int j = 0; j < 8; ++j) Bf[s][j] = *(const v2f*)(bp[j] + k);
  };
  auto MM = [&](int s) {
#pragma unroll
    for (int j = 0; j < 8; ++j) {
      acc[0][j] = wmma4(A0[s], Bf[s][j], acc[0][j]);
      acc[1][j] = wmma4(A1[s], Bf[s][j], acc[1][j]);
    }
  };

  LD(0, 0);
  LD(4, 1);
  int k = 0;
  for (; k + 24 <= K; k += 16) {       // steady state: lookahead of 2 steps
#pragma unroll
    for (int u = 0; u < 4; ++u) {
      LD(k + 8 + 4 * u, (u + 2) & 3);
      MM(u);
    }
  }
  // k == K-16 here (K % 16 == 0): 4 steps remain, slots 0,1 already loaded
  LD(k + 8, 2);  MM(0);
  LD(k + 12, 3); MM(1);
  MM(2);
  MM(3);

#pragma unroll
  for (int mh = 0; mh < 2; ++mh) {
#pragma unroll
    for (int i = 0; i < 8; ++i) {
      const int r = m_base + mh * 16 + half * 8 + i;
      if (r < cnt) {
        const int hr = plist ? plist[r] : r;
        float* orow = Hout + (size_t)hr * Ntot + n0 + m;
#pragma unroll
        for (int nt = 0; nt < 4; ++nt) {
          const float g = acc[mh][nt][i];
          const float u = acc[mh][nt + 4][i];
          orow[nt * 16] = (g / (1.f + __expf(-g))) * u;   // silu(g)*u
        }
      }
    }
  }
}

// -------------------------------------------------------------------------
// Down-projection GEMM, 32(M) x 128(N) tile per wave, 4-slot distance-2
// operand pipeline, one-wave workgroup with max register budget.
// pairs==nullptr: dense rows, output scaled by rowscale[t] (sigmoid shared
// gate) and written to Y (= d_out).  Else: gathered pair rows, unscaled.
// -------------------------------------------------------------------------
__global__ void __launch_bounds__(32, 1)
down_wmma(const float* __restrict__ Xh,   // [*, K]
          const float* __restrict__ W,    // [(e),N,K]
          float* __restrict__ Y,          // [*, N]
          const int* __restrict__ pairs,
          const int* __restrict__ counts,
          const float* __restrict__ rowscale,
          int K, int N)
{
  const int e   = blockIdx.z;
  const int cnt = pairs ? counts[e] : T_TOK;
  const int m_base = blockIdx.y * 32;
  if (m_base >= cnt) return;
  const int* plist = pairs ? (pairs + (size_t)e * T_TOK) : nullptr;
  const float* Wc  = W + (size_t)e * (size_t)N * (size_t)K;

  const int lane = threadIdx.x;
  const int m    = lane & 15;
  const int half = lane >> 4;
  const int kl   = half << 1;

  int gm0 = m_base + m;      if (gm0 > cnt - 1) gm0 = cnt - 1;
  int gm1 = m_base + 16 + m; if (gm1 > cnt - 1) gm1 = cnt - 1;
  const int xr0 = plist ? plist[gm0] : gm0;
  const int xr1 = plist ? plist[gm1] : gm1;
  const float* ap0 = Xh + (size_t)xr0 * K + kl;
  const float* ap1 = Xh + (size_t)xr1 * K + kl;

  const int n0 = blockIdx.x * 128;
  const float* bp[8];
#pragma unroll
  for (int nt = 0; nt < 8; ++nt)
    bp[nt] = Wc + (size_t)(n0 + nt * 16 + m) * K + kl;

  v8f acc[2][8];
  const v8f vzero = {};
#pragma unroll
  for (int mh = 0; mh < 2; ++mh)
#pragma unroll
    for (int j = 0; j < 8; ++j) acc[mh][j] = vzero;

  v2f A0[4], A1[4], Bf[4][8];

  auto LD = [&](int k, int s) {
    A0[s] = *(const v2f*)(ap0 + k);
    A1[s] = *(const v2f*)(ap1 + k);
#pragma unroll
    for (int j = 0; j < 8; ++j) Bf[s][j] = *(const v2f*)(bp[j] + k);
  };
  auto MM = [&](int s) {
#pragma unroll
    for (int j = 0; j < 8; ++j) {
      acc[0][j] = wmma4(A0[s], Bf[s][j], acc[0][j]);
      acc[1][j] = wmma4(A1[s], Bf[s][j], acc[1][j]);
    }
  };

  LD(0, 0);
  LD(4, 1);
  int k = 0;
  for (; k + 24 <= K; k += 16) {
#pragma unroll
    for (int u = 0; u < 4; ++u) {
      LD(k + 8 + 4 * u, (u + 2) & 3);
      MM(u);
    }
  }
  LD(k + 8, 2);  MM(0);
  LD(k + 12, 3); MM(1);
  MM(2);
  MM(3);

#pragma unroll
  for (int mh = 0; mh < 2; ++mh) {
#pragma unroll
    for (int i = 0; i < 8; ++i) {
      const int r = m_base + mh * 16 + half * 8 + i;
      if (r < cnt) {
        const int   yr = plist ? plist[r] : r;
        const float sc = rowscale ? rowscale[r] : 1.f;
        float* orow = Y + (size_t)yr * N + n0 + m;
#pragma unroll
        for (int nt = 0; nt < 8; ++nt) orow[nt * 16] = sc * acc[mh][nt][i];
      }
    }
  }
}

// out[t] += w0 * ypair[2t] + w1 * ypair[2t+1]   (out already holds shared path)
__global__ void combine_kernel(float* __restrict__ Out,
                               const float* __restrict__ Ypair,
                               const float* __restrict__ topw)
{
  const int idx = blockIdx.x * 256 + threadIdx.x;    // T*H total
  const int t = idx >> 11;                            // / H_DIM
  const int h = idx & (H_DIM - 1);
  const float w0 = topw[t * 2 + 0];
  const float w1 = topw[t * 2 + 1];
  Out[idx] += w0 * Ypair[(size_t)(t * 2 + 0) * H_DIM + h] +
              w1 * Ypair[(size_t)(t * 2 + 1) * H_DIM + h];
}

// -------------------------------------------------------------------------
extern "C" void kernel_launch(void* const* d_in, const int* in_sizes, int n_in,
                              void* d_out, int out_size, void* d_ws, size_t ws_size,
                              hipStream_t stream) {
  (void)in_sizes; (void)n_in; (void)out_size; (void)ws_size;
  const float* x   = (const float*)d_in[0];   // [T,H]
  const float* gw  = (const float*)d_in[1];   // [8,H]
  const float* w1  = (const float*)d_in[2];   // [8,2*I_MOE,H]
  const float* w2  = (const float*)d_in[3];   // [8,H,I_MOE]
  const float* sgu = (const float*)d_in[4];   // [2*I_SH,H]
  const float* sdw = (const float*)d_in[5];   // [H,I_SH]
  const float* sg  = (const float*)d_in[6];   // [1,H]
  float* out = (float*)d_out;                 // [T,H]

  char* ws = (char*)d_ws;
  size_t off = 0;
  auto carve = [&](size_t bytes) -> void* {
    void* p = ws + off;
    off = (off + bytes + 255) & ~((size_t)255);
    return p;
  };
  float* topw  = (float*)carve((size_t)T_TOK * 2 * 4);
  float* sgate = (float*)carve((size_t)T_TOK * 4);
  int*   cnts  = (int*)  carve((size_t)NEXP * 4);
  int*   pairs = (int*)  carve((size_t)NEXP * T_TOK * 4);
  float* h_sh  = (float*)carve((size_t)T_TOK * I_SH_ * 4);          // 23.1 MB
  float* h_moe = (float*)carve((size_t)T_TOK * 2 * I_MOE_ * 4);     // 11.5 MB
  float* ypair = (float*)carve((size_t)T_TOK * 2 * H_DIM * 4);      // 16.8 MB

  hipMemsetAsync(cnts, 0, NEXP * sizeof(int), stream);

  router_kernel<<<T_TOK, 256, 0, stream>>>(x, gw, sg, topw, sgate, cnts, pairs);

  // shared expert: gate_up (K=2048, N=5632) fused with silu*mul
  gateup_silu_wmma<<<dim3(I_SH_ / 64, T_TOK / 32, 1), 32, 0, stream>>>(
      x, sgu, h_sh, nullptr, nullptr, H_DIM, I_SH_);

  // routed experts: gate_up (K=2048, N=1408), gathered rows per expert
  gateup_silu_wmma<<<dim3(I_MOE_ / 64, T_TOK / 32, NEXP), 32, 0, stream>>>(
      x, w1, h_moe, pairs, cnts, H_DIM, I_MOE_);

  // shared down (K=5632, N=2048) -> d_out, fused sigmoid gate
  down_wmma<<<dim3(H_DIM / 128, T_TOK / 32, 1), 32, 0, stream>>>(
      h_sh, sdw, out, nullptr, nullptr, sgate, I_SH_, H_DIM);

  // expert down (K=1408, N=2048) -> per-pair rows
  down_wmma<<<dim3(H_DIM / 128, T_TOK / 32, NEXP), 32, 0, stream>>>(
      h_moe, w2, ypair, pairs, cnts, nullptr, I_MOE_, H_DIM);

  combine_kernel<<<(T_TOK * H_DIM) / 256, 256, 0, stream>>>(out, ypair, topw);
}